// AnalyticalDecoder_85581518340204
// MI455X (gfx1250) — compile-verified
//
#include <hip/hip_runtime.h>
#include <math.h>

typedef float v2f __attribute__((ext_vector_type(2)));
typedef float v8f __attribute__((ext_vector_type(8)));

#define RES    128
#define P_TOT  (RES * RES)        // 16384 grid points
#define G_TOT  1024               // BS*T Gaussians
#define GRP    16                 // Gaussians per workgroup (WMMA M)
#define NGRP   (G_TOT / GRP)      // 64
#define CHUNK  2048               // points per workgroup
#define NCHUNK (P_TOT / CHUNK)    // 8
#define WAVES  8
#define TPB    (WAVES * 32)

// ---------------------------------------------------------------------------
// Kernel 1: per-Gaussian coefficients of logp as a linear form over
// feat = [px^2, px*py, py^2, px, py, 1, 0, 0], plus min/max slot init.
// logp = -log(2pi) - 0.5*log(det) - 0.5*( d*dx^2 -(b+c)*dx*dy + a*dy^2 )/det
// ---------------------------------------------------------------------------
__global__ void coef_kernel(const float* __restrict__ mu,
                            const float* __restrict__ covar,
                            float* __restrict__ coef,
                            unsigned* __restrict__ wmin,
                            unsigned* __restrict__ wmax) {
    int g = blockIdx.x * blockDim.x + threadIdx.x;
    if (g >= G_TOT) return;
    float mux = mu[g * 2 + 0], muy = mu[g * 2 + 1];
    float a = covar[g * 4 + 0], b = covar[g * 4 + 1];
    float c = covar[g * 4 + 2], d = covar[g * 4 + 3];
    float det = a * d - b * c;
    float s   = -(b + c);
    float inv = -0.5f / det;
    float* cg = coef + g * 8;
    cg[0] = inv * d;                                   // px^2
    cg[1] = inv * s;                                   // px*py
    cg[2] = inv * a;                                   // py^2
    cg[3] = inv * (-2.f * d * mux - s * muy);          // px
    cg[4] = inv * (-2.f * a * muy - s * mux);          // py
    cg[5] = inv * (d * mux * mux + s * mux * muy + a * muy * muy)
            - 1.8378770664093453f                     // log(2*pi)
            - 0.5f * __logf(det);                      // constant
    cg[6] = 0.f;
    cg[7] = 0.f;
    wmin[g] = 0x7F800000u;   // +inf bits (probs are >= 0 -> uint-ordered)
    wmax[g] = 0u;            // 0.0f bits
}

// ---------------------------------------------------------------------------
// One 16(G) x 16(P) tile of logp via two fp32 WMMAs (K = 8 padded from 6).
// A (16x4, constant): lanes 0-15 hold M=lane, K={0,1}; lanes 16-31 K={2,3}.
// B (4x16, per-tile): VGPR0 lanes0-15=K0 / lanes16-31=K2; VGPR1 = K1/K3.
// Feature candidates computed unconditionally, then value-selects ->
// v_cndmask (no EXEC-masked branch in the hot loop; WMMA needs full EXEC).
// ---------------------------------------------------------------------------
__device__ __forceinline__ v8f tile_logp(v2f a0, v2f a1, int p, bool h) {
    const float step = 30.0f / 127.0f;
    float px = -15.f + (float)(p & (RES - 1)) * step;
    float py = -15.f + (float)(p >> 7) * step;
    float f0 = px * px;
    float f1 = px * py;
    float f2 = py * py;
    v2f b0, b1;
    b0.x = h ? f2  : f0;     // K2 : K0
    b0.y = h ? px  : f1;     // K3 : K1
    b1.x = h ? 0.f : py;     // K6 : K4
    b1.y = h ? 0.f : 1.f;    // K7 : K5
    v8f acc = {};
    acc = __builtin_amdgcn_wmma_f32_16x16x4_f32(false, a0, false, b0,
                                                (short)0, acc, false, false);
    acc = __builtin_amdgcn_wmma_f32_16x16x4_f32(false, a1, false, b1,
                                                (short)0, acc, false, false);
    return acc;
}

// ---------------------------------------------------------------------------
// Pass 1: per-Gaussian min/max over this workgroup's 2048 points.
// ---------------------------------------------------------------------------
__global__ void pass1_kernel(const float* __restrict__ coef,
                             unsigned* __restrict__ wmin,
                             unsigned* __restrict__ wmax) {
    const int g0   = blockIdx.x * GRP;
    const int tid  = threadIdx.x;
    const int wave = tid >> 5;
    const int lane = tid & 31;
    const int hi   = lane >> 4;
    const int ln   = lane & 15;
    const bool h   = hi != 0;

    const float* cg = coef + (size_t)(g0 + ln) * 8 + 2 * hi;
    v2f a0, a1;
    a0.x = cg[0]; a0.y = cg[1];      // K = 2*hi, 2*hi+1
    a1.x = cg[4]; a1.y = cg[5];      // K = 4+2*hi, 4+2*hi+1

    float rmin[8], rmax[8];
#pragma unroll
    for (int j = 0; j < 8; ++j) { rmin[j] = INFINITY; rmax[j] = 0.f; }

    const int p_start = blockIdx.y * CHUNK;
#pragma unroll 2
    for (int i = 0; i < CHUNK / (16 * WAVES); ++i) {
        int p = p_start + (i * WAVES + wave) * 16 + ln;
        v8f acc = tile_logp(a0, a1, p, h);
#pragma unroll
        for (int j = 0; j < 8; ++j) {
            float v = __expf(acc[j]);
            rmin[j] = fminf(rmin[j], v);
            rmax[j] = fmaxf(rmax[j], v);
        }
    }
    // Reduce over the 16 lanes of each half (same Gaussian row set per half).
#pragma unroll
    for (int off = 1; off < 16; off <<= 1) {
#pragma unroll
        for (int j = 0; j < 8; ++j) {
            rmin[j] = fminf(rmin[j], __shfl_xor(rmin[j], off, 32));
            rmax[j] = fmaxf(rmax[j], __shfl_xor(rmax[j], off, 32));
        }
    }
    if (ln == 0) {
#pragma unroll
        for (int j = 0; j < 8; ++j) {
            int g = g0 + j + 8 * hi;   // C/D row j -> M=j (lo half) / j+8 (hi)
            atomicMin(&wmin[g], __float_as_uint(rmin[j]));
            atomicMax(&wmax[g], __float_as_uint(rmax[j]));
        }
    }
}

// ---------------------------------------------------------------------------
// Pass 2: recompute (registers are cheaper than 128 MB of HBM traffic),
// normalize, single coalesced write of the 64 MB output.
// ---------------------------------------------------------------------------
__global__ void pass2_kernel(const float* __restrict__ coef,
                             const unsigned* __restrict__ wmin,
                             const unsigned* __restrict__ wmax,
                             float* __restrict__ out) {
    const int g0   = blockIdx.x * GRP;
    const int tid  = threadIdx.x;
    const int wave = tid >> 5;
    const int lane = tid & 31;
    const int hi   = lane >> 4;
    const int ln   = lane & 15;
    const bool h   = hi != 0;

    const float* cg = coef + (size_t)(g0 + ln) * 8 + 2 * hi;
    v2f a0, a1;
    a0.x = cg[0]; a0.y = cg[1];
    a1.x = cg[4]; a1.y = cg[5];

    float mn[8], sc[8];
#pragma unroll
    for (int j = 0; j < 8; ++j) {
        int g = g0 + j + 8 * hi;
        float lo = __uint_as_float(wmin[g]);
        float up = __uint_as_float(wmax[g]);
        mn[j] = lo;
        sc[j] = 1.0f / (up - lo);
    }

    const int p_start = blockIdx.y * CHUNK;
#pragma unroll 2
    for (int i = 0; i < CHUNK / (16 * WAVES); ++i) {
        int p = p_start + (i * WAVES + wave) * 16 + ln;
        v8f acc = tile_logp(a0, a1, p, h);
#pragma unroll
        for (int j = 0; j < 8; ++j) {
            float v = (__expf(acc[j]) - mn[j]) * sc[j];
            out[(size_t)(g0 + j + 8 * hi) * P_TOT + (size_t)p] = v;
        }
    }
}

// ---------------------------------------------------------------------------
extern "C" void kernel_launch(void* const* d_in, const int* in_sizes, int n_in,
                              void* d_out, int out_size, void* d_ws, size_t ws_size,
                              hipStream_t stream) {
    const float* mu    = (const float*)d_in[0];   // [16,64,2]
    const float* covar = (const float*)d_in[1];   // [16,64,2,2]
    float*    coef = (float*)d_ws;                                  // 32 KB
    unsigned* wmin = (unsigned*)((char*)d_ws + G_TOT * 8 * sizeof(float));
    unsigned* wmax = wmin + G_TOT;                                  // +8 KB
    float* out = (float*)d_out;                   // [16,64,1,128,128] fp32

    coef_kernel<<<(G_TOT + 255) / 256, 256, 0, stream>>>(mu, covar, coef, wmin, wmax);
    dim3 grid(NGRP, NCHUNK);   // 64 x 8 = 512 workgroups, 8 waves each
    pass1_kernel<<<grid, TPB, 0, stream>>>(coef, wmin, wmax);
    pass2_kernel<<<grid, TPB, 0, stream>>>(coef, wmin, wmax, out);
}